// HeteroPULLModel_86242943303867
// MI455X (gfx1250) — compile-verified
//
#include <hip/hip_runtime.h>

// Problem constants (match reference setup_inputs()).
#define NCV 50000   // compound nodes
#define NDV 50000   // disease nodes
#define FIN 256     // raw feature dim
#define HID 128     // hidden channels
#define OCH 64      // out channels
#define NE  500000  // edges per relation

typedef __attribute__((ext_vector_type(2))) float v2f;
typedef __attribute__((ext_vector_type(8))) float v8f;

// ---------------------------------------------------------------------------
// Pack W[K][NCOLS] (row-major) into per-lane WMMA-fragment order:
//   P[((ct*K/4 + kk/4)*32 + lane)] = { W[kk + 2*half][ct*16+mn],
//                                      W[kk + 2*half + 1][ct*16+mn] }
// so a wave's B fragment for k-step kk is ONE coalesced global_load_b64 at a
// compile-time immediate offset. Weights are tiny (<=128KB); repack per call.
// ---------------------------------------------------------------------------
template <int K, int NCOLS>
__global__ __launch_bounds__(256) void pack_w(const float* __restrict__ W,
                                              float2* __restrict__ P)
{
    constexpr int K4 = K / 4;
    constexpr int TOTAL = (NCOLS / 16) * K4 * 32;
    const int i = blockIdx.x * blockDim.x + threadIdx.x;
    if (i >= TOTAL) return;
    const int lane = i & 31;
    const int k4   = (i >> 5) % K4;
    const int ct   = (i >> 5) / K4;
    const int half = lane >> 4;
    const int mn   = lane & 15;
    const int col  = ct * 16 + mn;
    const int ka   = k4 * 4 + 2 * half;
    P[i] = make_float2(W[(size_t)ka * NCOLS + col],
                       W[(size_t)(ka + 1) * NCOLS + col]);
}

// ---------------------------------------------------------------------------
// Fused WMMA GEMM:  out = act( diag(1/max(cnt,1)) * A1 @ W1  [+ A2 @ W2] + b )
//   A1/A2 : [nrows, K] f32 row-major
//   P1/P2 : packed weights (see pack_w)
//   Block = 32 rows x NCOLS cols. Each wave owns 32 cols (2 col tiles) and
//   both 16-row tiles -> 4 accumulators; every A/B fragment feeds 2 v_wmma.
//   V_WMMA_F32_16X16X4_F32: full fp32 precision, matches the reference.
//   Row clamping on loads + guarded stores keep EXEC all-ones at every WMMA.
// ---------------------------------------------------------------------------
template <int K, int NCOLS, bool HAS2, bool RELU>
__global__ __launch_bounds__(256) void wmma_gemm(
    const float* __restrict__ A1, const float* __restrict__ cnt1,
    const float2* __restrict__ P1,
    const float* __restrict__ A2, const float2* __restrict__ P2,
    const float* __restrict__ bias,
    float* __restrict__ out, int nrows)
{
    constexpr int RB = 32;     // rows per block (two 16-row tiles)
    constexpr int KP = K + 4;  // pad: keeps fragment reads conflict-free, 16B align
    constexpr int K4 = K / 4;
    extern __shared__ float lds[];
    float* sA1 = lds;
    float* sA2 = HAS2 ? (lds + RB * KP) : nullptr;
    float* sScale = lds + (HAS2 ? 2 : 1) * RB * KP;

    const int t = threadIdx.x;
    const int row0 = blockIdx.x * RB;

    if (t < RB) {
        const int gr = min(row0 + t, nrows - 1);
        sScale[t] = cnt1 ? (1.0f / fmaxf(cnt1[gr], 1.0f)) : 1.0f;
    }
    __syncthreads();

    // Stage RBxK activation tile(s) as float4: b128 loads + b128 LDS stores.
    for (int i = t; i < RB * K4; i += blockDim.x) {
        const int r = i / K4, c = (i - r * K4) * 4;
        const int gr = min(row0 + r, nrows - 1);
        const float s = sScale[r];
        float4 v = *(const float4*)(A1 + (size_t)gr * K + c);
        v.x *= s; v.y *= s; v.z *= s; v.w *= s;
        *(float4*)(sA1 + r * KP + c) = v;
        if (HAS2)
            *(float4*)(sA2 + r * KP + c) = *(const float4*)(A2 + (size_t)gr * K + c);
    }
    __syncthreads();

    const int lane = t & 31;
    const int half = lane >> 4;   // K pair {2h,2h+1} / output rows M 8h..8h+7
    const int mn   = lane & 15;   // A row within tile / output column within tile
    const int ct0  = (t >> 5) * 2;  // first of this wave's two col tiles

    v8f acc[2][2];  // [rowTile][colTile]
#pragma unroll
    for (int i = 0; i < 2; ++i)
#pragma unroll
        for (int j = 0; j < 2; ++j)
            acc[i][j] = {0.f, 0.f, 0.f, 0.f, 0.f, 0.f, 0.f, 0.f};

    {
        const float2* Bp0 = P1 + (size_t)ct0 * K4 * 32 + lane;
        const float2* Bp1 = P1 + (size_t)(ct0 + 1) * K4 * 32 + lane;
        const float2* Ap0 = (const float2*)(sA1 + mn * KP + half * 2);
        const float2* Ap1 = (const float2*)(sA1 + (16 + mn) * KP + half * 2);
#pragma unroll
        for (int k4 = 0; k4 < K4; ++k4) {
            const float2 ta0 = Ap0[k4 * 2];
            const float2 ta1 = Ap1[k4 * 2];
            const float2 tb0 = Bp0[k4 * 32];
            const float2 tb1 = Bp1[k4 * 32];
            v2f a0, a1, b0, b1;
            a0.x = ta0.x; a0.y = ta0.y;  a1.x = ta1.x; a1.y = ta1.y;
            b0.x = tb0.x; b0.y = tb0.y;  b1.x = tb1.x; b1.y = tb1.y;
            acc[0][0] = __builtin_amdgcn_wmma_f32_16x16x4_f32(
                false, a0, false, b0, (short)0, acc[0][0], false, false);
            acc[0][1] = __builtin_amdgcn_wmma_f32_16x16x4_f32(
                false, a0, false, b1, (short)0, acc[0][1], false, false);
            acc[1][0] = __builtin_amdgcn_wmma_f32_16x16x4_f32(
                false, a1, false, b0, (short)0, acc[1][0], false, false);
            acc[1][1] = __builtin_amdgcn_wmma_f32_16x16x4_f32(
                false, a1, false, b1, (short)0, acc[1][1], false, false);
        }
    }
    if (HAS2) {
        const float2* Bp0 = P2 + (size_t)ct0 * K4 * 32 + lane;
        const float2* Bp1 = P2 + (size_t)(ct0 + 1) * K4 * 32 + lane;
        const float2* Ap0 = (const float2*)(sA2 + mn * KP + half * 2);
        const float2* Ap1 = (const float2*)(sA2 + (16 + mn) * KP + half * 2);
#pragma unroll
        for (int k4 = 0; k4 < K4; ++k4) {
            const float2 ta0 = Ap0[k4 * 2];
            const float2 ta1 = Ap1[k4 * 2];
            const float2 tb0 = Bp0[k4 * 32];
            const float2 tb1 = Bp1[k4 * 32];
            v2f a0, a1, b0, b1;
            a0.x = ta0.x; a0.y = ta0.y;  a1.x = ta1.x; a1.y = ta1.y;
            b0.x = tb0.x; b0.y = tb0.y;  b1.x = tb1.x; b1.y = tb1.y;
            acc[0][0] = __builtin_amdgcn_wmma_f32_16x16x4_f32(
                false, a0, false, b0, (short)0, acc[0][0], false, false);
            acc[0][1] = __builtin_amdgcn_wmma_f32_16x16x4_f32(
                false, a0, false, b1, (short)0, acc[0][1], false, false);
            acc[1][0] = __builtin_amdgcn_wmma_f32_16x16x4_f32(
                false, a1, false, b0, (short)0, acc[1][0], false, false);
            acc[1][1] = __builtin_amdgcn_wmma_f32_16x16x4_f32(
                false, a1, false, b1, (short)0, acc[1][1], false, false);
        }
    }

    // C/D layout: VGPR r -> M = r + 8*half, N = mn.
#pragma unroll
    for (int ct = 0; ct < 2; ++ct) {
        const int colBase = (ct0 + ct) * 16;
        const float bv = bias[colBase + mn];
#pragma unroll
        for (int tt = 0; tt < 2; ++tt) {
#pragma unroll
            for (int r = 0; r < 8; ++r) {
                const int row = row0 + tt * 16 + r + half * 8;
                float v = acc[tt][ct][r] + bv;
                if (RELU) v = fmaxf(v, 0.f);
                if (row < nrows)
                    out[(size_t)row * NCOLS + colBase + mn] = v;
            }
        }
    }
}

// ---------------------------------------------------------------------------
// Edge scatter for segment-mean numerator + counts.
// One wave = one edge: 32 lanes x float4 = one coalesced 512B source row.
// ---------------------------------------------------------------------------
__global__ __launch_bounds__(256) void scatter_mean_accum(
    const float* __restrict__ h, const int* __restrict__ ei,
    float* __restrict__ agg, float* __restrict__ cnt)
{
    const long long tid = (long long)blockIdx.x * blockDim.x + threadIdx.x;
    const long long total = (long long)NE * (HID / 4);
    if (tid >= total) return;
    const int e = (int)(tid >> 5);          // HID/4 == 32 lanes per edge
    const int q = ((int)tid & 31) << 2;     // feature quad
    const int src = ei[e];                  // ei row 0
    const int dst = ei[NE + e];             // ei row 1
    const float4 v = *reinterpret_cast<const float4*>(h + (size_t)src * HID + q);
    float* p = agg + (size_t)dst * HID + q;
    atomicAdd(p + 0, v.x);
    atomicAdd(p + 1, v.y);
    atomicAdd(p + 2, v.z);
    atomicAdd(p + 3, v.w);
    if (q == 0) atomicAdd(cnt + dst, 1.0f);
}

// ---------------------------------------------------------------------------
// Host side
// ---------------------------------------------------------------------------
static inline int cdiv(int a, int b) { return (a + b - 1) / b; }

static void graph_layer(const float* hc_in, const float* hd_in,
                        float* hc_out, float* hd_out,
                        float* aggC, float* aggD, float* cntC, float* cntD,
                        const float2* Prel_cd, const float* brel_cd, const float2* Proot_cd,
                        const float2* Prel_dc, const float* brel_dc, const float2* Proot_dc,
                        const int* ei_cd, const int* ei_dc, hipStream_t stream)
{
    hipMemsetAsync(aggD, 0, (size_t)NDV * HID * sizeof(float), stream);
    hipMemsetAsync(cntD, 0, (size_t)NDV * sizeof(float), stream);
    hipMemsetAsync(aggC, 0, (size_t)NCV * HID * sizeof(float), stream);
    hipMemsetAsync(cntC, 0, (size_t)NCV * sizeof(float), stream);

    const long long total = (long long)NE * (HID / 4);
    const int sgrid = (int)((total + 255) / 256);
    scatter_mean_accum<<<sgrid, 256, 0, stream>>>(hc_in, ei_cd, aggD, cntD);
    scatter_mean_accum<<<sgrid, 256, 0, stream>>>(hd_in, ei_dc, aggC, cntC);

    const size_t shL = (size_t)(2 * 32 * (HID + 4) + 32) * sizeof(float);
    // md = relu(mean-agg_d @ Wrel_cd + brel_cd + hd @ Wroot_cd)
    wmma_gemm<HID, HID, true, true><<<cdiv(NDV, 32), HID, shL, stream>>>(
        aggD, cntD, Prel_cd, hd_in, Proot_cd, brel_cd, hd_out, NDV);
    // mc = relu(mean-agg_c @ Wrel_dc + brel_dc + hc @ Wroot_dc)
    wmma_gemm<HID, HID, true, true><<<cdiv(NCV, 32), HID, shL, stream>>>(
        aggC, cntC, Prel_dc, hc_in, Proot_dc, brel_dc, hc_out, NCV);
}

extern "C" void kernel_launch(void* const* d_in, const int* in_sizes, int n_in,
                              void* d_out, int out_size, void* d_ws, size_t ws_size,
                              hipStream_t stream)
{
    (void)in_sizes; (void)n_in; (void)out_size; (void)ws_size;

    const float* x_c      = (const float*)d_in[0];
    const float* x_d      = (const float*)d_in[1];
    const float* W_in_c   = (const float*)d_in[2];
    const float* b_in_c   = (const float*)d_in[3];
    const float* W_in_d   = (const float*)d_in[4];
    const float* b_in_d   = (const float*)d_in[5];
    const float* Wrel0_cd = (const float*)d_in[6];
    const float* brel0_cd = (const float*)d_in[7];
    const float* Wroot0_cd= (const float*)d_in[8];
    const float* Wrel0_dc = (const float*)d_in[9];
    const float* brel0_dc = (const float*)d_in[10];
    const float* Wroot0_dc= (const float*)d_in[11];
    const float* Wrel1_cd = (const float*)d_in[12];
    const float* brel1_cd = (const float*)d_in[13];
    const float* Wroot1_cd= (const float*)d_in[14];
    const float* Wrel1_dc = (const float*)d_in[15];
    const float* brel1_dc = (const float*)d_in[16];
    const float* Wroot1_dc= (const float*)d_in[17];
    const float* W_out_c  = (const float*)d_in[18];
    const float* b_out_c  = (const float*)d_in[19];
    const float* W_out_d  = (const float*)d_in[20];
    const float* b_out_d  = (const float*)d_in[21];
    const int*   ei_cd    = (const int*)d_in[22];
    const int*   ei_dc    = (const int*)d_in[23];

    // Workspace carve-out (256B aligned).
    char* ws = (char*)d_ws;
    size_t off = 0;
    auto take = [&](size_t bytes) -> void* {
        void* p = ws + off;
        off = (off + bytes + 255) & ~(size_t)255;
        return p;
    };
    const size_t hbytes = (size_t)NCV * HID * sizeof(float);
    float* hcA  = (float*)take(hbytes);
    float* hdA  = (float*)take(hbytes);
    float* hcB  = (float*)take(hbytes);
    float* hdB  = (float*)take(hbytes);
    float* aggC = (float*)take(hbytes);
    float* aggD = (float*)take(hbytes);
    float* cntC = (float*)take((size_t)NCV * sizeof(float));
    float* cntD = (float*)take((size_t)NDV * sizeof(float));

    // Packed weight buffers.
    const size_t pin  = (size_t)FIN * HID / 2 * sizeof(float2);  // 256x128
    const size_t prr  = (size_t)HID * HID / 2 * sizeof(float2);  // 128x128
    const size_t pout = (size_t)HID * OCH / 2 * sizeof(float2);  // 128x64
    float2* Pin_c   = (float2*)take(pin);
    float2* Pin_d   = (float2*)take(pin);
    float2* Prel0cd = (float2*)take(prr);
    float2* Proot0cd= (float2*)take(prr);
    float2* Prel0dc = (float2*)take(prr);
    float2* Proot0dc= (float2*)take(prr);
    float2* Prel1cd = (float2*)take(prr);
    float2* Proot1cd= (float2*)take(prr);
    float2* Prel1dc = (float2*)take(prr);
    float2* Proot1dc= (float2*)take(prr);
    float2* Pout_c  = (float2*)take(pout);
    float2* Pout_d  = (float2*)take(pout);

    // 0) Repack weights into WMMA fragment order (cheap, <1MB total).
    {
        const int tIn  = (HID / 16) * (FIN / 4) * 32;
        const int tRR  = (HID / 16) * (HID / 4) * 32;
        const int tOut = (OCH / 16) * (HID / 4) * 32;
        pack_w<FIN, HID><<<cdiv(tIn, 256), 256, 0, stream>>>(W_in_c, Pin_c);
        pack_w<FIN, HID><<<cdiv(tIn, 256), 256, 0, stream>>>(W_in_d, Pin_d);
        pack_w<HID, HID><<<cdiv(tRR, 256), 256, 0, stream>>>(Wrel0_cd, Prel0cd);
        pack_w<HID, HID><<<cdiv(tRR, 256), 256, 0, stream>>>(Wroot0_cd, Proot0cd);
        pack_w<HID, HID><<<cdiv(tRR, 256), 256, 0, stream>>>(Wrel0_dc, Prel0dc);
        pack_w<HID, HID><<<cdiv(tRR, 256), 256, 0, stream>>>(Wroot0_dc, Proot0dc);
        pack_w<HID, HID><<<cdiv(tRR, 256), 256, 0, stream>>>(Wrel1_cd, Prel1cd);
        pack_w<HID, HID><<<cdiv(tRR, 256), 256, 0, stream>>>(Wroot1_cd, Proot1cd);
        pack_w<HID, HID><<<cdiv(tRR, 256), 256, 0, stream>>>(Wrel1_dc, Prel1dc);
        pack_w<HID, HID><<<cdiv(tRR, 256), 256, 0, stream>>>(Wroot1_dc, Proot1dc);
        pack_w<HID, OCH><<<cdiv(tOut, 256), 256, 0, stream>>>(W_out_c, Pout_c);
        pack_w<HID, OCH><<<cdiv(tOut, 256), 256, 0, stream>>>(W_out_d, Pout_d);
    }

    // 1) Input projections: h = relu(x @ W_in + b)   [K=256, NCOLS=128]
    const size_t shIn = (size_t)(32 * (FIN + 4) + 32) * sizeof(float);
    wmma_gemm<FIN, HID, false, true><<<cdiv(NCV, 32), HID, shIn, stream>>>(
        x_c, nullptr, Pin_c, nullptr, nullptr, b_in_c, hcA, NCV);
    wmma_gemm<FIN, HID, false, true><<<cdiv(NDV, 32), HID, shIn, stream>>>(
        x_d, nullptr, Pin_d, nullptr, nullptr, b_in_d, hdA, NDV);

    // 2) Two GraphConv layers (ping-pong A <-> B).
    graph_layer(hcA, hdA, hcB, hdB, aggC, aggD, cntC, cntD,
                Prel0cd, brel0_cd, Proot0cd, Prel0dc, brel0_dc, Proot0dc,
                ei_cd, ei_dc, stream);
    graph_layer(hcB, hdB, hcA, hdA, aggC, aggD, cntC, cntD,
                Prel1cd, brel1_cd, Proot1cd, Prel1dc, brel1_dc, Proot1dc,
                ei_cd, ei_dc, stream);

    // 3) Output projections, concatenated: [zc ; zd]  [K=128, NCOLS=64]
    float* out = (float*)d_out;
    const size_t shOut = (size_t)(32 * (HID + 4) + 32) * sizeof(float);
    wmma_gemm<HID, OCH, false, false><<<cdiv(NCV, 32), OCH, shOut, stream>>>(
        hcA, nullptr, Pout_c, nullptr, nullptr, b_out_c, out, NCV);
    wmma_gemm<HID, OCH, false, false><<<cdiv(NDV, 32), OCH, shOut, stream>>>(
        hdA, nullptr, Pout_d, nullptr, nullptr, b_out_d, out + (size_t)NCV * OCH, NDV);
}